// FramePrimerEncoder_1279900254348
// MI455X (gfx1250) — compile-verified
//
#include <hip/hip_runtime.h>
#include <math.h>

// FramePrimerEncoder for MI455X (gfx1250, wave32).
// All GEMMs use V_WMMA_F32_16X16X4_F32 (fp32-exact; kernel is HBM-bound at
// ~420MB mandatory traffic => ~18us @ 23.3TB/s, so fp32 matrix rate suffices).
// Softmax is fused into the A-panel load of the A*V GEMM to avoid a 128MB
// probability tensor. Transposed stores go through per-wave LDS tiles.
//
// Workspace layout (floats), peak ~48M + stats (~202 MB):
//   S0 [ 0M.. 8M): a_in  -> later qc
//   S1 [ 8M..16M): q     -> later attn
//   S2 [16M..24M): k     -> later t1 (low half)
//   S3 [24M..32M): v     -> later t1 (high half)
//   S4 [32M..40M): kc    -> later f_in
//   S5 [40M..48M): vc
//   [48M..): stats (m1,rinv1,s2,ss2,m2,rinv2, softmax max/rinv)

#define BQ 8
#define NB 4
#define HD 256
#define SZT (1024L*1024L)
#define EPSN 1e-5f

typedef float v2f __attribute__((ext_vector_type(2)));
typedef float v8f __attribute__((ext_vector_type(8)));

// ---------------- Kernel 1: in_norm + in_project + norm1 stats ----------------
__global__ __launch_bounds__(256) void k_norm_in(
    const float* __restrict__ x, const float* __restrict__ g_in,
    const float* __restrict__ b_in, const float* __restrict__ W_in,
    const float* __restrict__ bias_in, float* __restrict__ dout,
    float* __restrict__ st_m1, float* __restrict__ st_rinv1,
    float* __restrict__ st_s2, float* __restrict__ st_ss2) {
  int b = blockIdx.x >> 5;
  int w0 = (blockIdx.x & 31) << 5;
  int tx = threadIdx.x & 31, ty = threadIdx.x >> 5;
  const float* xb = x + (long)b * 2 * SZT;
  __shared__ float rs[8][32], rss[8][32], sm[32], sinv[32];
  float s = 0.f, ss = 0.f;
  for (int c = 0; c < 2; ++c) {
    const float* xc = xb + (long)c * SZT + w0 + tx;
    for (int i = ty; i < 1024; i += 8) {
      float v = xc[(long)i * 1024];
      s += v; ss += v * v;
    }
  }
  rs[ty][tx] = s; rss[ty][tx] = ss;
  __syncthreads();
  if (ty == 0) {
    float S = 0.f, SS = 0.f;
    for (int j = 0; j < 8; ++j) { S += rs[j][tx]; SS += rss[j][tx]; }
    float m = S * (1.f / 2048.f);
    float v = SS * (1.f / 2048.f) - m * m;
    sm[tx] = m; sinv[tx] = rsqrtf(v + EPSN);
  }
  __syncthreads();
  float m = sm[tx], inv = sinv[tx];
  float g = g_in[w0 + tx], bb = b_in[w0 + tx];
  float wc0 = W_in[0], wc1 = W_in[1], bi = bias_in[0];
  const float* x0p = xb + w0 + tx;
  const float* x1p = xb + SZT + w0 + tx;
  float* outp = dout + (long)(b * 3) * SZT + w0 + tx;
  float s2 = 0.f, ss2 = 0.f;
  for (int i = ty; i < 1024; i += 8) {
    float v0 = x0p[(long)i * 1024], v1 = x1p[(long)i * 1024];
    float n0 = (v0 - m) * inv * g + bb;
    float n1 = (v1 - m) * inv * g + bb;
    float x0v = n0 * wc0 + n1 * wc1 + bi;
    outp[(long)i * 1024] = x0v;
    s2 += x0v; ss2 += x0v * x0v;
  }
  rs[ty][tx] = s2; rss[ty][tx] = ss2;
  __syncthreads();
  if (ty == 0) {
    float S = 0.f, SS = 0.f;
    for (int j = 0; j < 8; ++j) { S += rs[j][tx]; SS += rss[j][tx]; }
    int idx = b * 1024 + w0 + tx;
    float m1 = S * (1.f / 1024.f);
    float v1 = SS * (1.f / 1024.f) - m1 * m1;
    st_m1[idx] = m1; st_rinv1[idx] = rsqrtf(v1 + EPSN);
    st_s2[idx] = S; st_ss2[idx] = SS;
  }
}

// ---------------- Kernel 2: norm1 + transpose -> a_in[b][w][h] ----------------
__global__ __launch_bounds__(256) void k_norm1_t(
    const float* __restrict__ dout, const float* __restrict__ st_m1,
    const float* __restrict__ st_rinv1, const float* __restrict__ g1,
    const float* __restrict__ b1, float* __restrict__ ain) {
  int b = blockIdx.x, w0 = blockIdx.y << 5, h0 = blockIdx.z << 5;
  int tx = threadIdx.x & 31, ty = threadIdx.x >> 5;
  __shared__ float T[32][33];
  const float* x0 = dout + (long)(b * 3) * SZT;
  for (int i = 0; i < 4; ++i)
    T[ty + 8 * i][tx] = x0[(long)(h0 + ty + 8 * i) * 1024 + w0 + tx];
  __syncthreads();
  for (int i = 0; i < 4; ++i) {
    int w = w0 + ty + 8 * i;
    int idx = b * 1024 + w;
    float m1 = st_m1[idx], r1 = st_rinv1[idx], g = g1[w], bb = b1[w];
    ain[((long)(b * 1024 + w)) * 1024 + h0 + tx] = (T[tx][ty + 8 * i] - m1) * r1 * g + bb;
  }
}

// ---------------- Kernel 3: depthwise conv1d (width 3, pad 1) along w ----------------
__global__ __launch_bounds__(256) void k_dwconv(
    const float* __restrict__ q, const float* __restrict__ k, const float* __restrict__ v,
    const float* __restrict__ kq, const float* __restrict__ kkw, const float* __restrict__ kv,
    float* __restrict__ qc, float* __restrict__ kc, float* __restrict__ vc) {
  int z = blockIdx.z;
  const float* src = (z == 0) ? q : (z == 1) ? k : v;
  const float* kw = (z == 0) ? kq : (z == 1) ? kkw : kv;
  float* dst = (z == 0) ? qc : (z == 1) ? kc : vc;
  long e = (long)blockIdx.x * 256 + threadIdx.x;
  int o = (int)(e & 1023);
  int w = (int)((e >> 10) & 1023);
  float l = (w > 0) ? src[e - 1024] : 0.f;
  float mid = src[e];
  float r = (w < 1023) ? src[e + 1024] : 0.f;
  dst[e] = kw[o * 3] * l + kw[o * 3 + 1] * mid + kw[o * 3 + 2] * r;
}

// ---------------- Kernel 4: softmax row stats (max, 1/sum) ----------------
__global__ __launch_bounds__(256) void k_softmax_stats(
    const float* __restrict__ qk, float* __restrict__ smax, float* __restrict__ srinv) {
  __shared__ float red[256];
  long r = blockIdx.x;
  const float* row = qk + r * 1024;
  int tid = threadIdx.x;
  float m = -3.4e38f;
  for (int i = tid; i < 1024; i += 256) m = fmaxf(m, row[i]);
  red[tid] = m; __syncthreads();
  for (int s = 128; s > 0; s >>= 1) {
    if (tid < s) red[tid] = fmaxf(red[tid], red[tid + s]);
    __syncthreads();
  }
  float M = red[0];
  __syncthreads();
  float s = 0.f;
  for (int i = tid; i < 1024; i += 256) s += __expf(row[i] - M);
  red[tid] = s; __syncthreads();
  for (int st = 128; st > 0; st >>= 1) {
    if (tid < st) red[tid] += red[tid + st];
    __syncthreads();
  }
  if (tid == 0) { smax[r] = M; srinv[r] = 1.f / red[0]; }
}

// ---------------- Kernel 5: norm2 stats (x0 sums reused + h4 sums) ----------------
__global__ __launch_bounds__(256) void k_h4_stats(
    const float* __restrict__ dout, const float* __restrict__ st_s2,
    const float* __restrict__ st_ss2, float* __restrict__ st_m2,
    float* __restrict__ st_rinv2) {
  int b = blockIdx.x >> 5;
  int w0 = (blockIdx.x & 31) << 5;
  int tx = threadIdx.x & 31, ty = threadIdx.x >> 5;
  const float* h4 = dout + (long)(b * 3 + 1) * SZT + w0 + tx;
  float s = 0.f, ss = 0.f;
  for (int i = ty; i < 1024; i += 8) {
    float v = h4[(long)i * 1024];
    s += v; ss += v * v;
  }
  __shared__ float rs[8][32], rss[8][32];
  rs[ty][tx] = s; rss[ty][tx] = ss;
  __syncthreads();
  if (ty == 0) {
    float S = 0.f, SS = 0.f;
    for (int j = 0; j < 8; ++j) { S += rs[j][tx]; SS += rss[j][tx]; }
    int idx = b * 1024 + w0 + tx;
    S += st_s2[idx]; SS += st_ss2[idx];
    float m = S * (1.f / 2048.f);
    float v = SS * (1.f / 2048.f) - m * m;
    st_m2[idx] = m; st_rinv2[idx] = rsqrtf(v + EPSN);
  }
}

// ---------------- Kernel 6: norm2 + ff_project + transpose -> f_in[b][w][h] ----------------
__global__ __launch_bounds__(256) void k_ffin_t(
    const float* __restrict__ dout, const float* __restrict__ st_m2,
    const float* __restrict__ st_rinv2, const float* __restrict__ g2,
    const float* __restrict__ b2, const float* __restrict__ W_ff,
    const float* __restrict__ bias_ff, float* __restrict__ fin) {
  int b = blockIdx.x, w0 = blockIdx.y << 5, h0 = blockIdx.z << 5;
  int tx = threadIdx.x & 31, ty = threadIdx.x >> 5;
  __shared__ float T0[32][33], T1[32][33];
  const float* x0 = dout + (long)(b * 3) * SZT;
  const float* h4 = dout + (long)(b * 3 + 1) * SZT;
  for (int i = 0; i < 4; ++i) {
    long off = (long)(h0 + ty + 8 * i) * 1024 + w0 + tx;
    T0[ty + 8 * i][tx] = x0[off];
    T1[ty + 8 * i][tx] = h4[off];
  }
  __syncthreads();
  float wf0 = W_ff[0], wf1 = W_ff[1], bf = bias_ff[0];
  for (int i = 0; i < 4; ++i) {
    int w = w0 + ty + 8 * i;
    int idx = b * 1024 + w;
    float m2 = st_m2[idx], r2 = st_rinv2[idx], g = g2[w], bb = b2[w];
    float n0 = (T0[tx][ty + 8 * i] - m2) * r2 * g + bb;
    float n1 = (T1[tx][ty + 8 * i] - m2) * r2 * g + bb;
    fin[((long)(b * 1024 + w)) * 1024 + h0 + tx] = n0 * wf0 + n1 * wf1 + bf;
  }
}

// ---------------- Templated WMMA GEMM:  C[m][n] = sum_k A[m][k] * B(k,n) ----------------
// BLAYOUT 0: B stored N x K (weight-style, B(k,n)=Bm[n*ldb+k])
// BLAYOUT 1: B stored K x N (B(k,n)=Bm[k*ldb+n])
// AMODE   1: A element -> exp(A - rowmax)*rowrinv  (fused softmax probs)
// EPI 0: plain store   1: relu^2 store   2: transposed store to Dz
// EPI 3: Dz = (acc + Dz)/32 + prev      4: transposed store to out2 channel `chan`
template <int BLAYOUT, int AMODE, int EPI>
__global__ __launch_bounds__(256) void gemm_wmma(
    const float* __restrict__ A, int lda, long az1, long az2,
    const float* __restrict__ Bm, int ldb, long bz1, long bz2,
    float* __restrict__ D, int ldd, long dz1, long dz2,
    int zdiv, int M, int N, int K,
    const float* __restrict__ rmax, const float* __restrict__ rsinv, int statStride,
    const float* __restrict__ prev, long pz1, long pz2,
    float* __restrict__ out2, int chan) {
  __shared__ float As[32][68];
  __shared__ float Bs[64][68];
  __shared__ float Ts[8][16 * 17];
  int z = blockIdx.z;
  int zq = z / zdiv, zr = z % zdiv;
  const float* Az = A + (long)zq * az1 + (long)zr * az2;
  const float* Bz = Bm + (long)zq * bz1 + (long)zr * bz2;
  float* Dz = D + (long)zq * dz1 + (long)zr * dz2;
  int m0 = blockIdx.y * 32;
  int n0 = blockIdx.x * 64;
  int tid = threadIdx.x;
  int wv = tid >> 5, lane = tid & 31, half = lane >> 4, l16 = lane & 15;
  int wm = wv >> 2, wn = wv & 3;
  v8f c = {0.f, 0.f, 0.f, 0.f, 0.f, 0.f, 0.f, 0.f};
  for (int k0 = 0; k0 < K; k0 += 64) {
    // Stage 32x64 A panel (k contiguous -> coalesced)
#pragma unroll
    for (int i = 0; i < 8; ++i) {
      int e = tid + 256 * i;
      int r = e >> 6, cc = e & 63;
      float v = Az[(long)(m0 + r) * lda + k0 + cc];
      if (AMODE == 1) {
        int gi = z * statStride + m0 + r;
        v = __expf(v - rmax[gi]) * rsinv[gi];
      }
      As[r][cc] = v;
    }
    // Stage 64x64 B panel
#pragma unroll
    for (int i = 0; i < 16; ++i) {
      int e = tid + 256 * i;
      if (BLAYOUT == 0) {
        int n = e >> 6, kk2 = e & 63;
        Bs[kk2][n] = Bz[(long)(n0 + n) * ldb + k0 + kk2];
      } else {
        int kk2 = e >> 6, n = e & 63;
        Bs[kk2][n] = Bz[(long)(k0 + kk2) * ldb + n0 + n];
      }
    }
    __syncthreads();
#pragma unroll
    for (int kk = 0; kk < 64; kk += 4) {
      v2f a, bf;
      // A 16x4 fragment layout: lane=M%16; vgpr0 K=2*(lane/16), vgpr1 K=+1
      a.x = As[wm * 16 + l16][kk + 2 * half];
      a.y = As[wm * 16 + l16][kk + 2 * half + 1];
      // B 4x16 fragment (mirror): lane=N%16; vgpr0 K=2*(lane/16), vgpr1 K=+1
      bf.x = Bs[kk + 2 * half][wn * 16 + l16];
      bf.y = Bs[kk + 2 * half + 1][wn * 16 + l16];
      c = __builtin_amdgcn_wmma_f32_16x16x4_f32(false, a, false, bf, (short)0, c,
                                                false, false);
    }
    __syncthreads();
  }
  int gm = m0 + wm * 16, gn = n0 + wn * 16;
  if (EPI == 0) {
#pragma unroll
    for (int j = 0; j < 8; ++j) {
      int r = j + 8 * half;  // C layout: vgpr j -> row j + 8*(lane/16), col lane%16
      Dz[(long)(gm + r) * ldd + gn + l16] = c[j];
    }
  } else if (EPI == 1) {
#pragma unroll
    for (int j = 0; j < 8; ++j) {
      int r = j + 8 * half;
      float t = fmaxf(c[j], 0.f);
      Dz[(long)(gm + r) * ldd + gn + l16] = t * t;
    }
  } else if (EPI == 3) {
    const float* Pz = prev + (long)zq * pz1 + (long)zr * pz2;
#pragma unroll
    for (int j = 0; j < 8; ++j) {
      int r = j + 8 * half;
      long idx = (long)(gm + r) * ldd + gn + l16;
      Dz[idx] = (c[j] + Dz[idx]) * 0.03125f + Pz[idx];  // 1/sqrt(1024)=1/32
    }
  } else {  // EPI 2 or 4: wave-local LDS transpose, then contiguous row stores
    float* ts = Ts[wv];
#pragma unroll
    for (int j = 0; j < 8; ++j) ts[(j + 8 * half) * 17 + l16] = c[j];
    // same-wave LDS ops are in-order; compiler inserts s_wait_dscnt before reads
#pragma unroll
    for (int j = 0; j < 8; ++j) {
      int r2 = j + 8 * half;  // transposed row = original col (n)
      float v = ts[l16 * 17 + r2];
      if (EPI == 2) {
        Dz[(long)(gn + r2) * ldd + gm + l16] = v;
      } else {
        int bb = gm >> 10;            // tiles never span batch (1024 % 16 == 0)
        int wb = (gm & 1023) + l16;   // original m -> w
        out2[((long)((bb * 3 + chan) * 1024 + gn + r2)) * 1024 + wb] = v;
      }
    }
  }
}

// ---------------- Launch ----------------
extern "C" void kernel_launch(void* const* d_in, const int* in_sizes, int n_in,
                              void* d_out, int out_size, void* d_ws, size_t ws_size,
                              hipStream_t stream) {
  const float* x       = (const float*)d_in[0];
  const float* prev_qk = (const float*)d_in[1];
  const float* g_in    = (const float*)d_in[2];
  const float* b_in    = (const float*)d_in[3];
  const float* W_in    = (const float*)d_in[4];
  const float* bias_in = (const float*)d_in[5];
  const float* g1      = (const float*)d_in[6];
  const float* b1      = (const float*)d_in[7];
  const float* Wq      = (const float*)d_in[8];
  const float* Wk      = (const float*)d_in[9];
  const float* Wv      = (const float*)d_in[10];
  const float* Wo      = (const float*)d_in[11];
  const float* kq      = (const float*)d_in[12];
  const float* kkw     = (const float*)d_in[13];
  const float* kv      = (const float*)d_in[14];
  const float* pos_w   = (const float*)d_in[15];
  const float* g2      = (const float*)d_in[16];
  const float* b2      = (const float*)d_in[17];
  const float* W_ff    = (const float*)d_in[18];
  const float* bias_ff = (const float*)d_in[19];
  const float* W1      = (const float*)d_in[20];
  const float* W2      = (const float*)d_in[21];

  float* dout  = (float*)d_out;                 // [B,3,1024,1024]
  float* qkout = dout + (long)BQ * 3 * SZT;     // [B,4,1024,1024]

  float* ws = (float*)d_ws;
  float* a_in = ws;                 // S0, later qc
  float* qb   = ws + 8 * SZT;       // S1, later attn
  float* kb   = ws + 16 * SZT;      // S2, later t1 (low)
  float* vb   = ws + 24 * SZT;      // S3, later t1 (high)
  float* kc   = ws + 32 * SZT;      // S4, later f_in
  float* vc   = ws + 40 * SZT;      // S5
  float* stats = ws + 48 * SZT;
  float* st_m1 = stats, *st_rinv1 = stats + 8192;
  float* st_s2 = stats + 16384, *st_ss2 = stats + 24576;
  float* st_m2 = stats + 32768, *st_rinv2 = stats + 40960;
  float* sm_max = stats + 49152, *sm_rinv = stats + 49152 + 32768;
  float* qc   = a_in;
  float* attn = qb;
  float* t1   = kb;   // 16M floats spanning S2+S3
  float* fin  = kc;

  // 1) in_norm + in_project -> out ch0; norm1 stats
  k_norm_in<<<256, 256, 0, stream>>>(x, g_in, b_in, W_in, bias_in, dout,
                                     st_m1, st_rinv1, st_s2, st_ss2);
  // 2) norm1 + transpose -> a_in[b][w][h]
  k_norm1_t<<<dim3(BQ, 32, 32), 256, 0, stream>>>(dout, st_m1, st_rinv1, g1, b1, a_in);

  // 3) QKV projections: [8192 x 1024] x Wx^T
  dim3 gqkv(1024 / 64, 8192 / 32, 1);
  gemm_wmma<0, 0, 0><<<gqkv, 256, 0, stream>>>(a_in, 1024, 0, 0, Wq, 1024, 0, 0,
      qb, 1024, 0, 0, 1, 8192, 1024, 1024, nullptr, nullptr, 0, nullptr, 0, 0, nullptr, 0);
  gemm_wmma<0, 0, 0><<<gqkv, 256, 0, stream>>>(a_in, 1024, 0, 0, Wk, 1024, 0, 0,
      kb, 1024, 0, 0, 1, 8192, 1024, 1024, nullptr, nullptr, 0, nullptr, 0, 0, nullptr, 0);
  gemm_wmma<0, 0, 0><<<gqkv, 256, 0, stream>>>(a_in, 1024, 0, 0, Wv, 1024, 0, 0,
      vb, 1024, 0, 0, 1, 8192, 1024, 1024, nullptr, nullptr, 0, nullptr, 0, 0, nullptr, 0);

  // 4) depthwise convs (q,k,v -> qc,kc,vc); q/k/v slots free after this
  k_dwconv<<<dim3(32768, 1, 3), 256, 0, stream>>>(qb, kb, vb, kq, kkw, kv, qc, kc, vc);

  // 5) Positional bias, transposed in place: qk[b,n,i,j] = sum_d Qc[j,d]*pos_w[d,i]
  gemm_wmma<1, 0, 2><<<dim3(16, 32, 32), 256, 0, stream>>>(
      qc, 1024, 1048576, 256, pos_w, 1024, 0, 0,
      qkout, 1024, 4194304, 1048576, 4, 1024, 1024, 256,
      nullptr, nullptr, 0, nullptr, 0, 0, nullptr, 0);
  // 6) qk = (Qc*Kc^T + P)/32 + prev_qk   (final qk output)
  gemm_wmma<0, 0, 3><<<dim3(16, 32, 32), 256, 0, stream>>>(
      qc, 1024, 1048576, 256, kc, 1024, 1048576, 256,
      qkout, 1024, 4194304, 1048576, 4, 1024, 1024, 256,
      nullptr, nullptr, 0, prev_qk, 4194304, 1048576, nullptr, 0);

  // 7) softmax row stats (max, 1/sum) over qk rows
  k_softmax_stats<<<32768, 256, 0, stream>>>(qkout, sm_max, sm_rinv);

  // 8) O = softmax(qk) * Vc, probs generated on the fly in the A-panel load
  gemm_wmma<1, 1, 0><<<dim3(4, 32, 32), 256, 0, stream>>>(
      qkout, 1024, 4194304, 1048576, vc, 1024, 1048576, 256,
      attn, 1024, 1048576, 256, 4, 1024, 256, 1024,
      sm_max, sm_rinv, 1024, nullptr, 0, 0, nullptr, 0);

  // 9) h4 = (attn * Wo^T) stored transposed into out ch1
  gemm_wmma<0, 0, 4><<<dim3(16, 256, 1), 256, 0, stream>>>(
      attn, 1024, 0, 0, Wo, 1024, 0, 0,
      dout, 1024, 0, 0, 1, 8192, 1024, 1024,
      nullptr, nullptr, 0, nullptr, 0, 0, dout, 1);

  // 10) norm2 stats (reuse x0 sums) ; 11) f_in[b][w][h]
  k_h4_stats<<<256, 256, 0, stream>>>(dout, st_s2, st_ss2, st_m2, st_rinv2);
  k_ffin_t<<<dim3(BQ, 32, 32), 256, 0, stream>>>(dout, st_m2, st_rinv2, g2, b2,
                                                 W_ff, bias_ff, fin);

  // 12) t1 = relu(f_in * W1^T)^2   [8192 x 2048]
  gemm_wmma<0, 0, 1><<<dim3(2048 / 64, 256, 1), 256, 0, stream>>>(
      fin, 1024, 0, 0, W1, 1024, 0, 0,
      t1, 2048, 0, 0, 1, 8192, 2048, 1024,
      nullptr, nullptr, 0, nullptr, 0, 0, nullptr, 0);
  // 13) hf = (t1 * W2^T) stored transposed into out ch2
  gemm_wmma<0, 0, 4><<<dim3(16, 256, 1), 256, 0, stream>>>(
      t1, 2048, 0, 0, W2, 2048, 0, 0,
      dout, 1024, 0, 0, 1, 8192, 1024, 2048,
      nullptr, nullptr, 0, nullptr, 0, 0, dout, 2);
}